// LatticeGaussian_62723702391546
// MI455X (gfx1250) — compile-verified
//
#include <hip/hip_runtime.h>
#include <hip/hip_bf16.h>

// LatticeGaussian exact kernel matvec: out = (W - I) @ U
//   W[i,j] = exp(-0.5 * ||ref_i - ref_j||^2),  N=8192, C=21, D=5
//
// Strategy: per-wave 16-row i-tile; j staged through LDS in 128-row chunks;
// W tile entries computed per-lane directly in the V_WMMA_F32_16X16X4_F32
// A-fragment layout; W@U accumulated with fp32 WMMA (two 16-wide channel
// tiles, C padded 21->32).

#define LG_N 8192
#define LG_C 21
#define LG_D 5
#define LG_CPAD 32
#define LG_JCHUNK 128

typedef __attribute__((ext_vector_type(2))) float v2f;
typedef __attribute__((ext_vector_type(8))) float v8f;

__global__ __launch_bounds__(256) void LatticeGaussian_62723702391546_kernel(
    const float* __restrict__ U,    // [N, C]
    const float* __restrict__ ref,  // [N, D]
    float* __restrict__ out)        // [N, C]
{
    __shared__ float sU[LG_JCHUNK * LG_CPAD];  // 16 KB, channels zero-padded
    __shared__ float sR[LG_JCHUNK * 8];        // 4 KB, ref rows padded to 8

    const int tid  = threadIdx.x;
    const int lane = tid & 31;
    const int wave = tid >> 5;
    const int h    = lane >> 4;      // half-wave: selects K pair / M offset
    const int n    = lane & 15;      // N index (channel within tile) / M row

    const int i0 = blockIdx.x * 128 + wave * 16;  // this wave's 16-row i-tile
    const int gm = i0 + n;                        // my A-matrix row (lanes 16-31 mirror)

    // My ref_i row (5 components) in registers.
    const float ri0 = ref[gm * LG_D + 0];
    const float ri1 = ref[gm * LG_D + 1];
    const float ri2 = ref[gm * LG_D + 2];
    const float ri3 = ref[gm * LG_D + 3];
    const float ri4 = ref[gm * LG_D + 4];

    v8f acc0 = {};  // channels 0..15
    v8f acc1 = {};  // channels 16..31 (21..31 are zero-padded)

    for (int j0 = 0; j0 < LG_N; j0 += LG_JCHUNK) {
        __syncthreads();  // protect previous chunk's consumers

        // Stage U chunk, zero-padded to 32 channels.
        for (int idx = tid; idx < LG_JCHUNK * LG_CPAD; idx += 256) {
            int row = idx >> 5;
            int c   = idx & 31;
            sU[idx] = (c < LG_C) ? U[(j0 + row) * LG_C + c] : 0.0f;
        }
        // Stage ref chunk, rows padded to stride 8.
        for (int idx = tid; idx < LG_JCHUNK * 8; idx += 256) {
            int row = idx >> 3;
            int d   = idx & 7;
            sR[idx] = (d < LG_D) ? ref[(j0 + row) * LG_D + d] : 0.0f;
        }
        __syncthreads();

        #pragma unroll
        for (int sub = 0; sub < LG_JCHUNK / 16; ++sub) {
            const int base = sub * 16 + 2 * h;  // this lane's K origin in the chunk
            #pragma unroll
            for (int s = 0; s < 4; ++s) {       // 4 x K4 steps = K16
                v2f a, b0, b1;
                #pragma unroll
                for (int v = 0; v < 2; ++v) {
                    const int jl = base + 4 * s + v;  // local j row
                    const float* r = &sR[jl * 8];
                    float dx0 = ri0 - r[0];
                    float dx1 = ri1 - r[1];
                    float dx2 = ri2 - r[2];
                    float dx3 = ri3 - r[3];
                    float dx4 = ri4 - r[4];
                    float d2 = dx0 * dx0 + dx1 * dx1 + dx2 * dx2 +
                               dx3 * dx3 + dx4 * dx4;
                    // exp(-0.5*d2) = exp2(-0.72134752*d2)  -> v_exp_f32
                    a[v]  = __builtin_amdgcn_exp2f(d2 * -0.72134752044448f);
                    b0[v] = sU[jl * LG_CPAD + n];
                    b1[v] = sU[jl * LG_CPAD + 16 + n];
                }
                acc0 = __builtin_amdgcn_wmma_f32_16x16x4_f32(
                    false, a, false, b0, (short)0, acc0, false, false);
                acc1 = __builtin_amdgcn_wmma_f32_16x16x4_f32(
                    false, a, false, b1, (short)0, acc1, false, false);
            }
        }
    }

    // Epilogue: D-matrix layout -> lane l, VGPR r holds (M = r + 8*h, N = n).
    #pragma unroll
    for (int r = 0; r < 8; ++r) {
        const int m  = r + 8 * h;
        const int gi = i0 + m;
        out[gi * LG_C + n] = acc0[r] - U[gi * LG_C + n];
        if (n < LG_C - 16) {  // channels 16..20
            out[gi * LG_C + 16 + n] = acc1[r] - U[gi * LG_C + 16 + n];
        }
    }
}

extern "C" void kernel_launch(void* const* d_in, const int* in_sizes, int n_in,
                              void* d_out, int out_size, void* d_ws, size_t ws_size,
                              hipStream_t stream) {
    (void)in_sizes; (void)n_in; (void)out_size; (void)d_ws; (void)ws_size;
    const float* U   = (const float*)d_in[0];
    const float* ref = (const float*)d_in[1];
    float* out       = (float*)d_out;

    dim3 grid(LG_N / 128);  // 64 workgroups, each covers 128 i-rows
    dim3 block(256);        // 8 waves of 32
    LatticeGaussian_62723702391546_kernel<<<grid, block, 0, stream>>>(U, ref, out);
}